// NCC_4544075399175
// MI455X (gfx1250) — compile-verified
//
#include <hip/hip_runtime.h>
#include <stdint.h>

// ---------------------------------------------------------------------------
// Local NCC loss, (2,1,160,192,160) fp32, win=21 (radius 10), eps=1e-5.
// Memory-bound separable box filter: 3 streaming passes + reduction.
// HBM traffic ~0.87 GB -> ~37us roofline at 23.3 TB/s on MI455X.
// Passes X/Y vectorized float4-per-lane along z -> b128 global accesses
// (512B per wave32 vmem instruction), 4x fewer vmem/VALU ops.
// CDNA5 paths: tensor_load_to_lds (TDM, HW LDS padding for conflict-free
// line stride), s_wait_tensorcnt, global_prefetch_b8.
// Workspace layout: bufA[5N] | bufB[5N] | partials[3840]  (~394 MB).
// ---------------------------------------------------------------------------

#define BDIM 2
#define XDIM 160
#define YDIM 192
#define ZDIM 160
#define NVOX (BDIM * XDIM * YDIM * ZDIM)  // 9,830,400
#define HALF 10                           // win = 21
#define NWIN 9261.0f                      // 21^3
#define NCC_EPS 1e-5f

static const size_t CH = (size_t)NVOX;    // channel stride in 5-channel buffers

typedef unsigned int u32;
typedef __attribute__((ext_vector_type(4))) unsigned int v4u;
typedef __attribute__((ext_vector_type(8))) int v8i;
typedef __attribute__((ext_vector_type(4))) int v4i;
typedef __attribute__((ext_vector_type(4))) float f4;

#if defined(__gfx1250__) && __has_builtin(__builtin_amdgcn_tensor_load_to_lds)
#define NCC_HAVE_TDM 1
#else
#define NCC_HAVE_TDM 0
#endif

__device__ __forceinline__ f4 ld4(const float* __restrict__ p, size_t o) {
  return *(const f4*)(p + o);
}
__device__ __forceinline__ void st4(float* __restrict__ p, size_t o, f4 v) {
  *(f4*)(p + o) = v;
}

// ---------------------------------------------------------------------------
// Pass 1: fuse products (I, J, I^2, J^2, IJ) + running box sum along X.
// Thread = one (b,y,z4) column, float4 per lane along z -> b128 coalesced.
// ---------------------------------------------------------------------------
__global__ __launch_bounds__(256) void ncc_pass_x(const float* __restrict__ I,
                                                  const float* __restrict__ J,
                                                  float* __restrict__ out) {
  const int ZQ = ZDIM / 4;  // 40 float4 groups per line
  const int t = blockIdx.x * blockDim.x + threadIdx.x;
  if (t >= BDIM * YDIM * ZQ) return;
  const int zq = t % ZQ;
  const int y = (t / ZQ) % YDIM;
  const int b = t / (ZQ * YDIM);
  const size_t xs = (size_t)YDIM * ZDIM;  // x stride (elements)
  const size_t colBase =
      (size_t)b * XDIM * xs + (size_t)y * ZDIM + (size_t)zq * 4;

  f4 s0 = {0, 0, 0, 0}, s1 = s0, s2 = s0, s3 = s0, s4 = s0;
  for (int x = 0; x <= HALF; ++x) {  // window for x=0 covers [0,10]
    const size_t o = colBase + (size_t)x * xs;
    const f4 i = ld4(I, o), j = ld4(J, o);
    s0 += i; s1 += j; s2 += i * i; s3 += j * j; s4 += i * j;
  }
  for (int x = 0; x < XDIM; ++x) {
    const size_t o = colBase + (size_t)x * xs;
    st4(out, o + 0 * CH, s0);
    st4(out, o + 1 * CH, s1);
    st4(out, o + 2 * CH, s2);
    st4(out, o + 3 * CH, s3);
    st4(out, o + 4 * CH, s4);
    const int xl = x + HALF + 1;            // leading edge enters window
    if (xl < XDIM) {
      const size_t ol = colBase + (size_t)xl * xs;
      const int xp = xl + 16;               // prefetch ahead of leading edge
      if (xp < XDIM) {
        __builtin_prefetch(&I[colBase + (size_t)xp * xs], 0, 1);
        __builtin_prefetch(&J[colBase + (size_t)xp * xs], 0, 1);
      }
      const f4 i = ld4(I, ol), j = ld4(J, ol);
      s0 += i; s1 += j; s2 += i * i; s3 += j * j; s4 += i * j;
    }
    const int xt = x - HALF;                // trailing edge leaves (L2 hit)
    if (xt >= 0) {
      const size_t ot = colBase + (size_t)xt * xs;
      const f4 i = ld4(I, ot), j = ld4(J, ot);
      s0 -= i; s1 -= j; s2 -= i * i; s3 -= j * j; s4 -= i * j;
    }
  }
}

// ---------------------------------------------------------------------------
// Pass 2: running box sum along Y for 5 channels, float4 per lane along z.
// ---------------------------------------------------------------------------
__global__ __launch_bounds__(256) void ncc_pass_y(const float* __restrict__ in,
                                                  float* __restrict__ out) {
  const int ZQ = ZDIM / 4;
  const int t = blockIdx.x * blockDim.x + threadIdx.x;
  if (t >= BDIM * XDIM * ZQ) return;
  const int zq = t % ZQ;
  const int x = (t / ZQ) % XDIM;
  const int b = t / (ZQ * XDIM);
  const size_t ys = (size_t)ZDIM;  // y stride (elements)
  const size_t colBase =
      (((size_t)b * XDIM + x) * YDIM) * ZDIM + (size_t)zq * 4;

  f4 s[5];
#pragma unroll
  for (int c = 0; c < 5; ++c) s[c] = (f4){0, 0, 0, 0};
  for (int y = 0; y <= HALF; ++y) {
    const size_t o = colBase + (size_t)y * ys;
#pragma unroll
    for (int c = 0; c < 5; ++c) s[c] += ld4(in, o + (size_t)c * CH);
  }
  for (int y = 0; y < YDIM; ++y) {
    const size_t o = colBase + (size_t)y * ys;
#pragma unroll
    for (int c = 0; c < 5; ++c) st4(out, o + (size_t)c * CH, s[c]);
    const int yl = y + HALF + 1;
    if (yl < YDIM) {
      const size_t ol = colBase + (size_t)yl * ys;
      const int yp = yl + 16;
      if (yp < YDIM) __builtin_prefetch(&in[colBase + (size_t)yp * ys], 0, 1);
#pragma unroll
      for (int c = 0; c < 5; ++c) s[c] += ld4(in, ol + (size_t)c * CH);
    }
    const int yt = y - HALF;
    if (yt >= 0) {
      const size_t ot = colBase + (size_t)yt * ys;
#pragma unroll
      for (int c = 0; c < 5; ++c) s[c] -= ld4(in, ot + (size_t)c * CH);
    }
  }
}

// ---------------------------------------------------------------------------
// Pass 3: TDM-stage 5ch x 16 z-lines x 160 into LDS, box sum along Z,
// per-voxel cc, block reduction -> partials[block].
// LDS layout uses TDM pad feature: +1 dword every 32 -> line stride 165 floats
// (165 mod 64 = 37, gcd(37,64)=1 -> bank-conflict-free line-parallel access).
// ---------------------------------------------------------------------------
#define TLINES 16
#define NSEG 16
#define SEGLEN (ZDIM / NSEG)     // 10
#define LELEM (TLINES * ZDIM)    // 2560 elements per channel per tile
#define CH_LDS 2640              // 2560 + 2560/32 pad dwords

__device__ __forceinline__ float ldsAt(const float* __restrict__ lds, int c,
                                       int line, int z) {
  return lds[c * CH_LDS + line * 165 + z + (z >> 5)];
}

__global__ __launch_bounds__(256) void ncc_pass_z(const float* __restrict__ in,
                                                  float* __restrict__ partials) {
  __shared__ float lds[5 * CH_LDS];  // 52,800 B
  __shared__ float red[256];
  const int wg = blockIdx.x;                         // 0..3839
  const size_t elemBase = (size_t)wg * LELEM;        // tile start in each channel

#if NCC_HAVE_TDM
  if (threadIdx.x == 0) {
    // ---- Tensor DMA descriptor: 2D tile 2560 x 5, dim0 stride = NVOX ----
    const uint64_t gaddr = (uint64_t)(uintptr_t)(in + elemBase);
    const u32 laddr = (u32)(uintptr_t)lds;  // low 32 bits = LDS byte offset
    v4u g0;
    g0.x = 1u;                                         // count=1, user desc
    g0.y = laddr;                                      // lds_addr [63:32]
    g0.z = (u32)gaddr;                                 // global_addr [95:64]
    g0.w = (u32)((gaddr >> 32) & 0x01FFFFFFu) | (2u << 30);  // addr hi | type=2
    v8i g1;
    g1[0] = (int)((2u << 16)      // data_size = 4B
                | (1u << 20)      // pad_enable
                | (4u << 22)      // pad_interval = 32 dwords
                | (0u << 25));    // pad_amount = 1 dword
    g1[1] = (int)(((u32)LELEM & 0xFFFFu) << 16);       // tensor_dim0 lo16
    g1[2] = (int)(((u32)LELEM >> 16) | (5u << 16));    // tensor_dim0 hi, dim1 lo
    g1[3] = (int)(((u32)LELEM) << 16);                 // dim1 hi=0, tile_dim0
    g1[4] = (int)(5u);                                 // tile_dim1=5, tile_dim2=0
    g1[5] = (int)((u32)NVOX);                          // tensor_dim0_stride lo32
    g1[6] = 0;                                         // stride0 hi, stride1 lo
    g1[7] = 0;                                         // stride1 hi
    v4i gz4 = {0, 0, 0, 0};                            // groups 2/3 unused (2D)
    v8i gz8 = {0, 0, 0, 0, 0, 0, 0, 0};                // 6-arg form extra group
    __builtin_amdgcn_tensor_load_to_lds(g0, g1, gz4, gz4, gz8, 0);
    __builtin_amdgcn_s_wait_tensorcnt(0);
  }
  __syncthreads();
#else
  // Fallback: cooperative coalesced load into the same padded layout.
  for (int i = threadIdx.x; i < 5 * LELEM; i += 256) {
    const int c = i / LELEM;
    const int r = i % LELEM;
    lds[c * CH_LDS + r + (r >> 5)] = in[(size_t)c * CH + elemBase + r];
  }
  __syncthreads();
#endif

  // ---- running box sum along z + cc ----
  const int line = threadIdx.x & (TLINES - 1);  // 0..15
  const int seg = threadIdx.x >> 4;             // 0..15
  const int z0 = seg * SEGLEN;

  float s[5] = {0.f, 0.f, 0.f, 0.f, 0.f};
  for (int z = z0 - HALF; z <= z0 + HALF; ++z) {
    if (z >= 0 && z < ZDIM) {
#pragma unroll
      for (int c = 0; c < 5; ++c) s[c] += ldsAt(lds, c, line, z);
    }
  }

  const float invN = 1.0f / NWIN;
  float acc = 0.f;
#pragma unroll
  for (int k = 0; k < SEGLEN; ++k) {
    const int z = z0 + k;
    // s0=I_sum s1=J_sum s2=I2_sum s3=J2_sum s4=IJ_sum
    const float cross = s[4] - s[0] * s[1] * invN;
    const float Ivar = s[2] - s[0] * s[0] * invN;
    const float Jvar = s[3] - s[1] * s[1] * invN;
    acc += (cross * cross) / (Ivar * Jvar + NCC_EPS);
    const int zl = z + HALF + 1;
    if (zl < ZDIM) {
#pragma unroll
      for (int c = 0; c < 5; ++c) s[c] += ldsAt(lds, c, line, zl);
    }
    const int zt = z - HALF;
    if (zt >= 0) {
#pragma unroll
      for (int c = 0; c < 5; ++c) s[c] -= ldsAt(lds, c, line, zt);
    }
  }

  red[threadIdx.x] = acc;
  __syncthreads();
  for (int w = 128; w > 0; w >>= 1) {
    if ((int)threadIdx.x < w) red[threadIdx.x] += red[threadIdx.x + w];
    __syncthreads();
  }
  if (threadIdx.x == 0) partials[wg] = red[0];
}

// ---------------------------------------------------------------------------
// Final: deterministic fixed-order sum of partials -> 1 - mean(cc).
// ---------------------------------------------------------------------------
__global__ __launch_bounds__(256) void ncc_finalize(const float* __restrict__ partials,
                                                    int np, float* __restrict__ out) {
  __shared__ float red[256];
  float a = 0.f;
  for (int i = threadIdx.x; i < np; i += 256) a += partials[i];
  red[threadIdx.x] = a;
  __syncthreads();
  for (int w = 128; w > 0; w >>= 1) {
    if ((int)threadIdx.x < w) red[threadIdx.x] += red[threadIdx.x + w];
    __syncthreads();
  }
  if (threadIdx.x == 0) out[0] = 1.0f - red[0] / (float)NVOX;
}

// ---------------------------------------------------------------------------
extern "C" void kernel_launch(void* const* d_in, const int* in_sizes, int n_in,
                              void* d_out, int out_size, void* d_ws, size_t ws_size,
                              hipStream_t stream) {
  (void)in_sizes; (void)n_in; (void)out_size; (void)ws_size;
  const float* I = (const float*)d_in[0];
  const float* J = (const float*)d_in[1];
  float* bufA = (float*)d_ws;                 // 5 * NVOX floats
  float* bufB = bufA + (size_t)5 * NVOX;      // 5 * NVOX floats
  float* partials = bufB + (size_t)5 * NVOX;  // 3840 floats

  const int colsX = BDIM * YDIM * (ZDIM / 4);  // 15360 -> 60 blocks
  const int colsY = BDIM * XDIM * (ZDIM / 4);  // 12800 -> 50 blocks
  const int tilesZ = (BDIM * XDIM * YDIM) / TLINES;  // 3840

  ncc_pass_x<<<(colsX + 255) / 256, 256, 0, stream>>>(I, J, bufA);
  ncc_pass_y<<<(colsY + 255) / 256, 256, 0, stream>>>(bufA, bufB);
  ncc_pass_z<<<tilesZ, 256, 0, stream>>>(bufB, partials);
  ncc_finalize<<<1, 256, 0, stream>>>(partials, tilesZ, (float*)d_out);
}